// QuantumNeuralNetwork_6004364280361
// MI455X (gfx1250) — compile-verified
//
#include <hip/hip_runtime.h>

#define NQ 12
#define DIM 4096          // 2^NQ amplitudes
#define NL 6
#define NC 64
#define BATCH 512
#define THREADS 256
#define PER_T (DIM / THREADS)   // 16 amplitudes per thread

typedef __attribute__((ext_vector_type(2))) float v2f;
typedef __attribute__((ext_vector_type(8))) float v8f;

// ---------------------------------------------------------------------------
// Kernel 1: state-vector simulation, one sample per workgroup.
// State double-buffered in LDS (64 KB). wire 0 = MSB -> bit (NQ-1-w).
// The per-layer ring of 12 CNOTs is a fixed basis permutation; we apply it
// as ONE gather pass per layer using precomputed per-thread source indices.
// ---------------------------------------------------------------------------
__global__ __launch_bounds__(THREADS) void qsim_kernel(
    const float* __restrict__ x,      // (BATCH, NQ)
    const float* __restrict__ qw,     // (NL, NQ)
    float* __restrict__ z_out)        // (BATCH, NQ)
{
  __shared__ float smem[4 * DIM];          // two complex buffers (64 KB)
  __shared__ float th[NQ * (NL + 1)];      // 84 gate angles

  const int b = blockIdx.x;
  const int t = threadIdx.x;

  float* reC = smem;                // current buffer
  float* imC = smem + DIM;
  float* reO = smem + 2 * DIM;      // other buffer
  float* imO = smem + 3 * DIM;

  // Preload angles: th[0..11] = x[b,:], th[12..83] = qweights flat
  if (t < NQ)      th[t] = x[b * NQ + t];
  if (t < NL * NQ) th[NQ + t] = qw[t];

  // |0...0>
  for (int j = t; j < DIM; j += THREADS) {
    reC[j] = (j == 0) ? 1.0f : 0.0f;
    imC[j] = 0.0f;
  }

  // Precompute CNOT-ring gather sources: new[i] = old[G(i)],
  // G = f_1∘f_2∘...∘f_12 (apply last gate's index map first).
  int src[PER_T];
  #pragma unroll
  for (int k = 0; k < PER_T; ++k) {
    int j = t + k * THREADS;
    #pragma unroll
    for (int w = NQ - 1; w >= 0; --w) {
      const int cbit = 1 << (NQ - 1 - w);
      const int tbit = 1 << (NQ - 1 - ((w + 1) % NQ));
      if (j & cbit) j ^= tbit;
    }
    src[k] = j;
  }
  __syncthreads();

  // RX(theta) on wire w, in place on current buffer:
  //   re0' = c*re0 + s*im1 ; im0' = c*im0 - s*re1
  //   re1' = c*re1 + s*im0 ; im1' = c*im1 - s*re0
  auto apply_rx = [&](int w, float theta) {
    float s, c;
    sincosf(theta * 0.5f, &s, &c);
    const int bit = 1 << (NQ - 1 - w);
    const int low = bit - 1;
    #pragma unroll 4
    for (int p = t; p < DIM / 2; p += THREADS) {
      const int i0 = ((p & ~low) << 1) | (p & low);
      const int i1 = i0 | bit;
      const float r0 = reC[i0], m0 = imC[i0];
      const float r1 = reC[i1], m1 = imC[i1];
      reC[i0] = c * r0 + s * m1;
      imC[i0] = c * m0 - s * r1;
      reC[i1] = c * r1 + s * m0;
      imC[i1] = c * m1 - s * r0;
    }
    __syncthreads();
  };

  // AngleEmbedding
  for (int w = 0; w < NQ; ++w) apply_rx(w, th[w]);

  // BasicEntanglerLayers: 12 RX then one fused CNOT-ring permutation
  for (int l = 0; l < NL; ++l) {
    for (int w = 0; w < NQ; ++w) apply_rx(w, th[NQ + l * NQ + w]);
    #pragma unroll
    for (int k = 0; k < PER_T; ++k) {
      const int i = t + k * THREADS;
      reO[i] = reC[src[k]];
      imO[i] = imC[src[k]];
    }
    __syncthreads();
    float* tr = reC; reC = reO; reO = tr;
    float* ti = imC; imC = imO; imO = ti;
  }

  // <Z_w> = sum_i p_i * (+1 if bit(w)==0 else -1)
  float acc[NQ];
  #pragma unroll
  for (int w = 0; w < NQ; ++w) acc[w] = 0.0f;
  #pragma unroll
  for (int k = 0; k < PER_T; ++k) {
    const int i = t + k * THREADS;
    const float p = reC[i] * reC[i] + imC[i] * imC[i];
    #pragma unroll
    for (int w = 0; w < NQ; ++w)
      acc[w] += ((i >> (NQ - 1 - w)) & 1) ? -p : p;
  }
  // partials go to the *other* buffer -> no extra barrier needed before write
  #pragma unroll
  for (int w = 0; w < NQ; ++w) reO[w * THREADS + t] = acc[w];
  __syncthreads();
  for (int stride = THREADS / 2; stride > 0; stride >>= 1) {
    if (t < stride) {
      #pragma unroll
      for (int w = 0; w < NQ; ++w)
        reO[w * THREADS + t] += reO[w * THREADS + t + stride];
    }
    __syncthreads();
  }
  if (t < NQ) z_out[b * NQ + t] = reO[t * THREADS];
}

// ---------------------------------------------------------------------------
// Kernel 2: out(512x64) = z(512x12) @ W^T(12x64) + b via V_WMMA_F32_16X16X4_F32
// One wave32 per 16x16 output tile; K=12 -> 3 WMMA steps of K=4.
// A: lanes 0-15 M=lane K={k0,k0+1}; lanes 16-31 same M, K={k0+2,k0+3}.
// B: lanes 0-15 N=lane K={k0,k0+1}; lanes 16-31 same N, K={k0+2,k0+3}.
// C/D: VGPR r -> (M=r, N=lane) lanes 0-15; (M=8+r, N=lane-16) lanes 16-31.
// ---------------------------------------------------------------------------
__global__ __launch_bounds__(128) void classifier_wmma_kernel(
    const float* __restrict__ z,      // (BATCH, NQ)
    const float* __restrict__ W,      // (NC, NQ)  -> B[k][n] = W[n][k]
    const float* __restrict__ bias,   // (NC,)
    float* __restrict__ out)          // (BATCH, NC)
{
  const int lane = threadIdx.x & 31;
  const int wave = threadIdx.x >> 5;
  const int gw   = blockIdx.x * 4 + wave;   // 0..127
  const int m0   = (gw >> 2) * 16;          // 32 M tiles
  const int n0   = (gw & 3) * 16;           // 4  N tiles

  const int half = lane >> 4;               // 0: lanes 0-15, 1: lanes 16-31
  const int l15  = lane & 15;
  const int n    = n0 + l15;
  const int mRow = m0 + l15;

  v8f c;
  const float bn = bias[n];
  #pragma unroll
  for (int r = 0; r < 8; ++r) c[r] = bn;    // D = A*B + bias

  #pragma unroll
  for (int k0 = 0; k0 < NQ; k0 += 4) {
    const int kA = k0 + half * 2;
    v2f a, bm;
    a.x  = z[mRow * NQ + kA];
    a.y  = z[mRow * NQ + kA + 1];
    bm.x = W[n * NQ + kA];
    bm.y = W[n * NQ + kA + 1];
    c = __builtin_amdgcn_wmma_f32_16x16x4_f32(
        false, a, false, bm, (short)0, c, false, false);
  }

  const int mOut = m0 + half * 8;
  #pragma unroll
  for (int r = 0; r < 8; ++r)
    out[(mOut + r) * NC + n] = c[r];
}

// ---------------------------------------------------------------------------
extern "C" void kernel_launch(void* const* d_in, const int* in_sizes, int n_in,
                              void* d_out, int out_size, void* d_ws, size_t ws_size,
                              hipStream_t stream) {
  const float* x  = (const float*)d_in[0];   // (512, 12)
  const float* qw = (const float*)d_in[1];   // (6, 12)
  const float* W  = (const float*)d_in[2];   // (64, 12)
  const float* b  = (const float*)d_in[3];   // (64,)
  float* z = (float*)d_ws;                   // (512, 12) scratch

  qsim_kernel<<<BATCH, THREADS, 0, stream>>>(x, qw, z);
  classifier_wmma_kernel<<<32, 128, 0, stream>>>(z, W, b, (float*)d_out);
}